// RoiAlign_19576460935212
// MI455X (gfx1250) — compile-verified
//
#include <hip/hip_runtime.h>
#include <hip/hip_bf16.h>
#include <stdint.h>

// Problem shape (fixed by the reference harness).
#define FM_N 4
#define FM_C 256
#define FM_H 200
#define FM_W 200
#define GH 14
#define GW 14
#define GSZ (GH * GW)        // 196 samples per ROI per channel
#define HALF_ELEMS 7168      // floats per staging buffer (28 KB); max window 82*82=6724 fits

#if defined(__AMDGCN__) && __has_builtin(__builtin_amdgcn_tensor_load_to_lds)
#define USE_TDM 1
#else
#define USE_TDM 0
#endif

typedef unsigned int u32v4 __attribute__((ext_vector_type(4)));
typedef int          i32v4 __attribute__((ext_vector_type(4)));
typedef int          i32v8 __attribute__((ext_vector_type(8)));

#if USE_TDM
__device__ __forceinline__ int rfl(int v) { return __builtin_amdgcn_readfirstlane(v); }

// Issue one TDM load: a [win_w x win_h x nch] f32 tile of the feature map
// (strides 1 / FM_W / FM_H*FM_W elements) into LDS at lds_byte_addr.
// D# bit layout per CDNA5 ISA.
__device__ __forceinline__ void tdm_load_window(const float* fm, int b, int c0, int nch,
                                                int rmin, int cmin, int win_w, int win_h,
                                                unsigned lds_byte_addr) {
  b = rfl(b); c0 = rfl(c0); nch = rfl(nch);
  rmin = rfl(rmin); cmin = rfl(cmin);
  win_w = rfl(win_w); win_h = rfl(win_h);
  lds_byte_addr = (unsigned)rfl((int)lds_byte_addr);

  uint64_t gaddr = (uint64_t)(uintptr_t)(fm +
      ((((size_t)b * FM_C + (size_t)c0) * FM_H + (size_t)rmin) * FM_W + (size_t)cmin));

  // Group 0: count=1 | lds_addr | global_addr[56:0] | type=2
  u32v4 G0;
  G0[0] = 1u;
  G0[1] = lds_byte_addr;
  G0[2] = (unsigned)(gaddr & 0xFFFFFFFFull);
  G0[3] = (unsigned)((gaddr >> 32) & 0x01FFFFFFull) | (2u << 30);

  unsigned td0 = (unsigned)win_w;         // tensor_dim0 (elements)
  unsigned td1 = (unsigned)win_h;         // tensor_dim1
  unsigned st0 = (unsigned)FM_W;          // tensor_dim0_stride = row stride (elements)
  unsigned st1 = (unsigned)(FM_H * FM_W); // tensor_dim1_stride = channel stride (elements)

  // Group 1: mask=0, data_size=4B(code 2), no barrier/iterate/pad;
  // tensor_dim0 @ [79:48], tensor_dim1 @ [111:80], tile_dim0 @ [127:112],
  // tile_dim1 @ [143:128], tile_dim2 @ [159:144], stride0 @ [207:160], stride1 @ [255:208].
  i32v8 G1;
  G1[0] = (int)(2u << 16);
  G1[1] = (int)((td0 & 0xFFFFu) << 16);
  G1[2] = (int)(((td0 >> 16) & 0xFFFFu) | ((td1 & 0xFFFFu) << 16));
  G1[3] = (int)(((td1 >> 16) & 0xFFFFu) | (((unsigned)win_w & 0xFFFFu) << 16));
  G1[4] = (int)(((unsigned)win_h & 0xFFFFu) | (((unsigned)nch & 0xFFFFu) << 16));
  G1[5] = (int)st0;
  G1[6] = (int)((st1 & 0xFFFFu) << 16);
  G1[7] = (int)(st1 >> 16);

  // Group 2: tensor_dim2 = nch; no dim3, iterate disabled.
  i32v4 G2; G2[0] = nch; G2[1] = 0; G2[2] = 0; G2[3] = 0;
  i32v4 G3 = {0, 0, 0, 0};

#if __clang_major__ >= 23
  i32v8 GX = {0, 0, 0, 0, 0, 0, 0, 0};
  __builtin_amdgcn_tensor_load_to_lds(G0, G1, G2, G3, GX, 0);
#else
  __builtin_amdgcn_tensor_load_to_lds(G0, G1, G2, G3, 0);
#endif
}
#endif // USE_TDM

__global__ __launch_bounds__(256) void roialign_tdm_kernel(
    const float* __restrict__ fm, const float* __restrict__ rois,
    const int* __restrict__ img_h, float* __restrict__ out) {

  __shared__ float s_win[2][HALF_ELEMS];          // double-buffered TDM staging (2 x 28 KB)
  __shared__ int   s_ry0[GH], s_ry1[GH], s_rx0[GW], s_rx1[GW]; // window-relative corners
  __shared__ float s_ly[GH], s_lx[GW];
  __shared__ int   s_my[GH], s_mx[GW];
  __shared__ int   s_meta[8];                     // b, rmin, cmin, win_w, win_h, cpt

  const int tid = threadIdx.x;
  const int m = blockIdx.x;

  // ---- Phase 1: per-ROI interpolation tables (trivial scalar work) ----
  if (tid == 0) {
    const float inv_stride = (float)FM_H / (float)(*img_h);  // H / img = 1/stride
    const float* r = rois + (size_t)m * 6;
    int   b   = (int)r[0];
    float bx1 = r[2] * inv_stride, by1 = r[3] * inv_stride;
    float bx2 = r[4] * inv_stride, by2 = r[5] * inv_stride;
    int rmin = FM_H - 1, rmax = 0, cmin = FM_W - 1, cmax = 0;
    for (int i = 0; i < GH; ++i) {
      float t  = (float)i * (1.0f / (float)(GH - 1));
      float ys = by1 + (by2 - by1) * t;
      float xs = bx1 + (bx2 - bx1) * t;
      int myok = (ys >= 0.0f) && (ys <= (float)(FM_H - 1));
      int mxok = (xs >= 0.0f) && (xs <= (float)(FM_W - 1));
      float yf = floorf(ys), xf = floorf(xs);
      int y0  = (int)fminf(fmaxf(yf,        0.0f), (float)(FM_H - 1));
      int y1c = (int)fminf(fmaxf(yf + 1.0f, 0.0f), (float)(FM_H - 1));
      int x0  = (int)fminf(fmaxf(xf,        0.0f), (float)(FM_W - 1));
      int x1c = (int)fminf(fmaxf(xf + 1.0f, 0.0f), (float)(FM_W - 1));
      s_ly[i] = ys - yf;  s_lx[i] = xs - xf;
      s_my[i] = myok;     s_mx[i] = mxok;
      s_ry0[i] = y0;  s_ry1[i] = y1c;  s_rx0[i] = x0;  s_rx1[i] = x1c;
      rmin = min(rmin, y0);  rmax = max(rmax, y1c);
      cmin = min(cmin, x0);  cmax = max(cmax, x1c);
    }
    for (int i = 0; i < GH; ++i) {
      s_ry0[i] -= rmin; s_ry1[i] -= rmin; s_rx0[i] -= cmin; s_rx1[i] -= cmin;
    }
    int win_w = cmax - cmin + 1, win_h = rmax - rmin + 1;
    int winsize = win_w * win_h;                  // <= 82*82 = 6724 <= HALF_ELEMS
    int cpt = HALF_ELEMS / winsize;               // channels per staging tile
    if (cpt > FM_C) cpt = FM_C;
    if (cpt < 1) cpt = 1;
    s_meta[0] = b; s_meta[1] = rmin; s_meta[2] = cmin;
    s_meta[3] = win_w; s_meta[4] = win_h; s_meta[5] = cpt;
  }
  __syncthreads();

  const int b     = s_meta[0];
  const int rmin  = s_meta[1];
  const int cmin  = s_meta[2];
  const int win_w = s_meta[3];
  const int win_h = s_meta[4];
  const int cpt   = s_meta[5];
  const int winsize = win_w * win_h;
  const int ntile = (FM_C + cpt - 1) / cpt;

#if USE_TDM
  // Prime the pipeline: wave 0 launches the DMA for tile 0.
  if (tid < 32) {
    int nch0 = (cpt < FM_C) ? cpt : FM_C;
    tdm_load_window(fm, b, 0, nch0, rmin, cmin, win_w, win_h,
                    (unsigned)(uintptr_t)&s_win[0][0]);
  }
#endif

  for (int t = 0; t < ntile; ++t) {
    const int c0  = t * cpt;
    const int nch = min(cpt, FM_C - c0);

#if USE_TDM
    if (tid < 32) {
      if (t + 1 < ntile) {
        // Prefetch next tile into the other buffer, then wait for the current one.
        const int c0n  = c0 + cpt;
        const int nchn = min(cpt, FM_C - c0n);
        tdm_load_window(fm, b, c0n, nchn, rmin, cmin, win_w, win_h,
                        (unsigned)(uintptr_t)&s_win[(t + 1) & 1][0]);
        __builtin_amdgcn_s_wait_tensorcnt(1);  // TDM ops complete in order per wave
      } else {
        __builtin_amdgcn_s_wait_tensorcnt(0);
      }
    }
#else
    // Fallback: cooperative coalesced copy of the window into LDS.
    {
      const float* src = fm + (((size_t)b * FM_C + c0) * FM_H + rmin) * FM_W + cmin;
      const int total = nch * winsize;
      for (int q = tid; q < total; q += 256) {
        int cl = q / winsize, rem = q - cl * winsize;
        int rr = rem / win_w, cc = rem - rr * win_w;
        s_win[t & 1][q] = src[((size_t)cl * FM_H + rr) * FM_W + cc];
      }
    }
#endif
    __syncthreads();

    // ---- Sample 14x14 bilinear taps for this channel tile out of LDS ----
    // Process horizontally adjacent (j, j+1) pairs: shared y row bases and ly,
    // one coalesced 8-byte store per pair (p even -> 8B aligned).
    const float* buf = &s_win[t & 1][0];
    const int work2 = nch * (GSZ / 2);  // pairs per tile
    for (int w = tid; w < work2; w += 256) {
      int cl = w / (GSZ / 2);
      int q  = w - cl * (GSZ / 2);      // 0..97
      int i  = q / (GW / 2);            // row 0..13
      int jp = (q - i * (GW / 2)) * 2;  // j = jp, jp+1 (same row)

      const float ly = s_ly[i];
      const int row0 = cl * winsize + s_ry0[i] * win_w;
      const int row1 = cl * winsize + s_ry1[i] * win_w;
      const int okY  = s_my[i];

      float2 v2;
      {
        const int x0 = s_rx0[jp], x1 = s_rx1[jp];
        const float lx = s_lx[jp];
        float g00 = buf[row0 + x0], g01 = buf[row0 + x1];
        float g10 = buf[row1 + x0], g11 = buf[row1 + x1];
        float top = g00 + lx * (g01 - g00);
        float bot = g10 + lx * (g11 - g10);
        float v = top + ly * (bot - top);
        v2.x = (okY & s_mx[jp]) ? v : 0.0f;
      }
      {
        const int x0 = s_rx0[jp + 1], x1 = s_rx1[jp + 1];
        const float lx = s_lx[jp + 1];
        float g00 = buf[row0 + x0], g01 = buf[row0 + x1];
        float g10 = buf[row1 + x0], g11 = buf[row1 + x1];
        float top = g00 + lx * (g01 - g00);
        float bot = g10 + lx * (g11 - g10);
        float v = top + ly * (bot - top);
        v2.y = (okY & s_mx[jp + 1]) ? v : 0.0f;
      }
      // out[m][c][i][j]: consecutive threads -> consecutive pairs -> coalesced b64 stores
      *(float2*)&out[((size_t)m * FM_C + (size_t)(c0 + cl)) * GSZ + i * GW + jp] = v2;
    }
    __syncthreads();  // all readers done before the next DMA overwrites this buffer
  }
}

extern "C" void kernel_launch(void* const* d_in, const int* in_sizes, int n_in,
                              void* d_out, int out_size, void* d_ws, size_t ws_size,
                              hipStream_t stream) {
  const float* fm   = (const float*)d_in[0];  // [4, 256, 200, 200] f32
  const float* rois = (const float*)d_in[1];  // [M, 6] f32
  const int*   imgh = (const int*)d_in[2];    // scalar 800
  float* out = (float*)d_out;                 // [M, 256, 14, 14] f32
  const int M = in_sizes[1] / 6;              // 2048
  roialign_tdm_kernel<<<dim3(M), dim3(256), 0, stream>>>(fm, rois, imgh, out);
}